// GravityAEModel_38929583571368
// MI455X (gfx1250) — compile-verified
//
#include <hip/hip_runtime.h>
#include <hip/hip_bf16.h>
#include <math.h>
#include <stdint.h>

typedef __attribute__((ext_vector_type(2))) float v2f;
typedef __attribute__((ext_vector_type(8))) float v8f;

#define IN_DIM   128
#define HID_DIM  128
#define OC_DIM   65      // OUT + 1
#define OCP_DIM  80      // padded to 5 * 16
#define EPSV     1e-5f

// ---------------------------------------------------------------------------
// Hardware fp32 atomic add, device scope (cross-WGP coherent; ISA: RMW at
// scope <= cache-scope is performed locally, so SCOPE_DEV is required here).
// ---------------------------------------------------------------------------
__device__ __forceinline__ void atomic_add_f32_dev(float* p, float v) {
    asm volatile("global_atomic_add_f32 %0, %1, off scope:SCOPE_DEV"
                 :: "v"(p), "v"(v) : "memory");
}

// Async copy of 16 bytes global -> LDS (ASYNCcnt-tracked).
__device__ __forceinline__ void async_copy_b128(uint32_t lds_off, const void* gaddr) {
    asm volatile("global_load_async_to_lds_b128 %0, %1, off"
                 :: "v"(lds_off), "v"(gaddr) : "memory");
}

__device__ __forceinline__ void wait_asynccnt0() {
    asm volatile("s_wait_asynccnt 0x0" ::: "memory");
}

// ---------------------------------------------------------------------------
// Degree / normalization
// ---------------------------------------------------------------------------
__global__ void deg_init_kernel(float* dinv, int n) {
    int i = blockIdx.x * blockDim.x + threadIdx.x;
    if (i < n) dinv[i] = 1.0f;               // self-loop contributes 1
}

__global__ void deg_edge_kernel(float* dinv, const int* dst, int e) {
    int i = blockIdx.x * blockDim.x + threadIdx.x;
    if (i < e) atomic_add_f32_dev(&dinv[dst[i]], 1.0f);
}

__global__ void deg_fin_kernel(float* dinv, int n) {
    int i = blockIdx.x * blockDim.x + threadIdx.x;
    if (i < n) dinv[i] = rsqrtf(dinv[i]);    // deg >= 1 always
}

// ---------------------------------------------------------------------------
// fp32 WMMA GEMM:  C[16*gridDim.x x 16*nwaves] = A (MxK, lda) * B (KxN, ldb)
// K must be 128.  One wave -> one 16x16 tile; the block first stages the
// 16x128 A-strip into LDS with async global->LDS copies, so the 5-8 waves
// of the block read A from LDS instead of 5-8x redundant global loads.
// Requires M % 16 == 0 and full blocks (EXEC all ones for WMMA).
// ---------------------------------------------------------------------------
__global__ void wmma_gemm_f32_kernel(const float* __restrict__ A,
                                     const float* __restrict__ B,
                                     float* __restrict__ C,
                                     int K, int lda, int ldb, int ldc) {
    __shared__ __align__(16) float shA[16 * 128];   // 8 KB strip

    const int lane = threadIdx.x & 31;
    const int wave = threadIdx.x >> 5;
    const int lo   = lane & 15;     // M (for A) / N (for B) within tile
    const int hi   = lane >> 4;     // selects K pair {0,1} vs {2,3}
    const int rowBase = blockIdx.x << 4;
    const int colBase = wave << 4;

    // --- stage A strip: 16 rows x 128 cols = 512 x 16B chunks ---
    const uint32_t shBase = (uint32_t)(uintptr_t)shA;   // low 32 bits = LDS offset
    for (int c = threadIdx.x; c < 512; c += blockDim.x) {
        int row  = c >> 5;          // /32 chunks per row
        int col4 = c & 31;          // 4-float chunk within row
        async_copy_b128(shBase + (uint32_t)(((row << 7) + (col4 << 2)) << 2),
                        A + (size_t)(rowBase + row) * lda + (col4 << 2));
    }
    wait_asynccnt0();
    __syncthreads();

    const float* bp = B + (size_t)(hi << 1) * ldb + colBase + lo;
    const float* arow = shA + (lo << 7) + (hi << 1);

    v8f acc = {};
    for (int k = 0; k < K; k += 4) {
        v2f a = *(const v2f*)(arow + k);     // ds_load_b64
        v2f b;
        b.x = bp[0];                         // B row k+2*hi
        b.y = bp[ldb];                       // B row k+2*hi+1
        bp += (size_t)4 * ldb;
        acc = __builtin_amdgcn_wmma_f32_16x16x4_f32(
                  false, a, false, b, (short)0, acc, false, false);
    }

    float* cp = C + (size_t)(rowBase + (hi << 3)) * ldc + colBase + lo;
#pragma unroll
    for (int r = 0; r < 8; ++r)
        cp[(size_t)r * ldc] = acc[r];        // C: VGPR r -> row r (lanes 0-15), r+8 (16-31)
}

// ---------------------------------------------------------------------------
// Pad W2 (128 x 65) -> W2p (128 x 80), zeros in the tail columns
// ---------------------------------------------------------------------------
__global__ void pad_w2_kernel(const float* __restrict__ W2, float* __restrict__ W2p) {
    int idx = blockIdx.x * blockDim.x + threadIdx.x;
    if (idx >= HID_DIM * OCP_DIM) return;
    int r = idx / OCP_DIM, c = idx % OCP_DIM;
    W2p[idx] = (c < OC_DIM) ? W2[r * OC_DIM + c] : 0.0f;
}

// ---------------------------------------------------------------------------
// Aggregation: out[i,c] = in[i,c] * dinv[i]^2  (self-loop term, also resets state)
// ---------------------------------------------------------------------------
__global__ void agg_init_kernel(const float* __restrict__ in, float* __restrict__ out,
                                const float* __restrict__ dinv, int n, int ncols, int ld) {
    int idx = blockIdx.x * blockDim.x + threadIdx.x;
    if (idx >= n * ncols) return;
    int i = idx / ncols, c = idx % ncols;
    float di = dinv[i];
    out[(size_t)i * ld + c] = in[(size_t)i * ld + c] * di * di;
}

// Scatter: out[dst,c] += in[src,c] * dinv[src]*dinv[dst].  One wave per edge.
__global__ void agg_edge_kernel(const float* __restrict__ in, float* __restrict__ out,
                                const float* __restrict__ dinv,
                                const int* __restrict__ src, const int* __restrict__ dst,
                                int ncols, int ld, int e) {
    int eid = (blockIdx.x * blockDim.x + threadIdx.x) >> 5;
    if (eid >= e) return;
    int lane = threadIdx.x & 31;
    int s = src[eid], d = dst[eid];
    float norm = dinv[s] * dinv[d];
    const float* ip = in + (size_t)s * ld;
    float*       op = out + (size_t)d * ld;
    for (int c = lane; c < ncols; c += 32)
        atomic_add_f32_dev(&op[c], ip[c] * norm);
}

// ---------------------------------------------------------------------------
// v = leaky( bn( v + bias ) ), in place
// ---------------------------------------------------------------------------
__global__ void bn_act_kernel(float* buf, const float* __restrict__ bias,
                              const float* __restrict__ gamma, const float* __restrict__ beta,
                              const float* __restrict__ mean,  const float* __restrict__ var,
                              int n, int ncols, int ld) {
    int idx = blockIdx.x * blockDim.x + threadIdx.x;
    if (idx >= n * ncols) return;
    int i = idx / ncols, c = idx % ncols;
    size_t p = (size_t)i * ld + c;
    float v = buf[p] + bias[c];
    v = (v - mean[c]) * rsqrtf(var[c] + EPSV) * gamma[c] + beta[c];
    buf[p] = (v >= 0.0f) ? v : 0.1f * v;
}

// ---------------------------------------------------------------------------
// Per-edge output: sigmoid( mj[dst] - || pos[src] - pos[dst] || )
// z has ld = 80; cols 0..63 = pos, col 64 = mj.  One wave per edge.
// ---------------------------------------------------------------------------
__global__ void edge_out_kernel(const float* __restrict__ z,
                                const int* __restrict__ src, const int* __restrict__ dst,
                                float* __restrict__ out, int e) {
    int eid = (blockIdx.x * blockDim.x + threadIdx.x) >> 5;
    if (eid >= e) return;
    int lane = threadIdx.x & 31;
    int s = src[eid], d = dst[eid];
    const float* ps = z + (size_t)s * OCP_DIM;
    const float* pd = z + (size_t)d * OCP_DIM;
    v2f a = *(const v2f*)(ps + (lane << 1));
    v2f b = *(const v2f*)(pd + (lane << 1));
    float dx = a.x - b.x, dy = a.y - b.y;
    float sum = dx * dx + dy * dy;
#pragma unroll
    for (int off = 16; off > 0; off >>= 1)
        sum += __shfl_xor(sum, off);
    if (lane == 0) {
        float t = pd[64] - sqrtf(sum);
        out[eid] = 1.0f / (1.0f + expf(-t));
    }
}

// ---------------------------------------------------------------------------
extern "C" void kernel_launch(void* const* d_in, const int* in_sizes, int n_in,
                              void* d_out, int out_size, void* d_ws, size_t ws_size,
                              hipStream_t stream) {
    const float* x      = (const float*)d_in[0];
    const int*   ei     = (const int*)  d_in[1];
    const float* W1     = (const float*)d_in[2];
    const float* b1     = (const float*)d_in[3];
    const float* gamma1 = (const float*)d_in[4];
    const float* beta1  = (const float*)d_in[5];
    const float* mean1  = (const float*)d_in[6];
    const float* var1   = (const float*)d_in[7];
    const float* W2     = (const float*)d_in[8];
    const float* b2     = (const float*)d_in[9];
    const float* gamma2 = (const float*)d_in[10];
    const float* beta2  = (const float*)d_in[11];
    const float* mean2  = (const float*)d_in[12];
    const float* var2   = (const float*)d_in[13];

    const int n = in_sizes[0] / IN_DIM;      // 50000 (multiple of 16)
    const int e = in_sizes[1] / 2;           // 800000
    const int* src = ei;
    const int* dst = ei + e;
    float* out = (float*)d_out;

    // workspace layout (floats)
    float* ws   = (float*)d_ws;
    float* xw   = ws;                          // n*128 (later reused as zw, ld=80)
    float* agg1 = xw   + (size_t)n * HID_DIM;  // n*128
    float* agg2 = agg1 + (size_t)n * HID_DIM;  // n*80
    float* dinv = agg2 + (size_t)n * OCP_DIM;  // n
    float* W2p  = dinv + n;                    // 128*80

    const int TB = 256;
    const int rowTiles  = n / 16;              // 3125
    const int edgeWaves = (e * 32 + TB - 1) / TB;

    // --- degree normalization ---
    deg_init_kernel<<<(n + TB - 1) / TB, TB, 0, stream>>>(dinv, n);
    deg_edge_kernel<<<(e + TB - 1) / TB, TB, 0, stream>>>(dinv, dst, e);
    deg_fin_kernel<<<(n + TB - 1) / TB, TB, 0, stream>>>(dinv, n);

    // --- pad W2 ---
    pad_w2_kernel<<<(HID_DIM * OCP_DIM + TB - 1) / TB, TB, 0, stream>>>(W2, W2p);

    // --- layer 1: xw = x @ W1 (WMMA fp32), aggregate, bn+leaky ---
    wmma_gemm_f32_kernel<<<rowTiles, 256, 0, stream>>>(x, W1, xw,
                                                       IN_DIM, IN_DIM, HID_DIM, HID_DIM);
    agg_init_kernel<<<(n * HID_DIM + TB - 1) / TB, TB, 0, stream>>>(xw, agg1, dinv,
                                                                    n, HID_DIM, HID_DIM);
    agg_edge_kernel<<<edgeWaves, TB, 0, stream>>>(xw, agg1, dinv, src, dst,
                                                  HID_DIM, HID_DIM, e);
    bn_act_kernel<<<(n * HID_DIM + TB - 1) / TB, TB, 0, stream>>>(agg1, b1, gamma1, beta1,
                                                                  mean1, var1, n, HID_DIM, HID_DIM);

    // --- layer 2: zw = h @ W2p (WMMA fp32, 5 col tiles), aggregate, bn+leaky ---
    float* zw = xw;                            // reuse, ld = 80
    wmma_gemm_f32_kernel<<<rowTiles, 160, 0, stream>>>(agg1, W2p, zw,
                                                       HID_DIM, HID_DIM, OCP_DIM, OCP_DIM);
    agg_init_kernel<<<(n * OC_DIM + TB - 1) / TB, TB, 0, stream>>>(zw, agg2, dinv,
                                                                   n, OC_DIM, OCP_DIM);
    agg_edge_kernel<<<edgeWaves, TB, 0, stream>>>(zw, agg2, dinv, src, dst,
                                                  OC_DIM, OCP_DIM, e);
    bn_act_kernel<<<(n * OC_DIM + TB - 1) / TB, TB, 0, stream>>>(agg2, b2, gamma2, beta2,
                                                                 mean2, var2, n, OC_DIM, OCP_DIM);

    // --- edge readout ---
    edge_out_kernel<<<edgeWaves, TB, 0, stream>>>(agg2, src, dst, out, e);
}